// MultiHeadAttention_22582938042661
// MI455X (gfx1250) — compile-verified
//
#include <hip/hip_runtime.h>
#include <hip/hip_bf16.h>

typedef __attribute__((ext_vector_type(2))) float v2f;
typedef __attribute__((ext_vector_type(8))) float v8f;

#define SEQ   1024
#define NB    8        // batch
#define NH    8        // heads
#define DKH   64       // d_k per head
#define IND   512      // model dim
#define ROWS  (NB*SEQ) // 8192 rows for projections

#define TPAD  517      // LDS row stride (floats) for the 16x512 K/V tile

// ---------------------------------------------------------------------------
// gfx1250 async copy: global -> LDS, 16B per lane, tracked by ASYNCcnt.
// ---------------------------------------------------------------------------
__device__ __forceinline__ void async_ld_b128(unsigned lds_byte_addr,
                                              const float* gaddr) {
    asm volatile("global_load_async_to_lds_b128 %0, %1, off"
                 :: "v"(lds_byte_addr), "v"(gaddr) : "memory");
}
__device__ __forceinline__ void wait_async0() {
    asm volatile("s_wait_asynccnt 0" ::: "memory");
}

// Stage one 16-row x 512-col f32 tile (gsrc, row-major, contiguous) into LDS
// at ldsThread-derived addresses (row stride TPAD floats). 256 threads x 8
// b128 chunks each = 32KB. Caller must wait_async0() + __syncthreads().
__device__ __forceinline__ void stage_tile_async(const float* gsrc,
                                                 unsigned ldsThreadBase) {
    const float* g = gsrc + (threadIdx.x << 2);       // 4 floats per chunk
#pragma unroll
    for (int j = 0; j < 8; ++j) {
        // chunk c = tid + 256*j : row advances by 2 per j (128 chunks/row)
        async_ld_b128(ldsThreadBase + (unsigned)(j * 2 * TPAD * 4),
                      g + (j << 10));
    }
}

// ---------------------------------------------------------------------------
// Kernel 1: Q/K/V projections.  Y[row,col] = sum_c X[row,c]*W[col,c] + b[col]
// One wave per 16x16 output tile, K stepped by 4 via v_wmma_f32_16x16x4_f32.
// ---------------------------------------------------------------------------
__global__ void mha_proj_kernel(const float* __restrict__ x,
                                const float* __restrict__ wq, const float* __restrict__ bq,
                                const float* __restrict__ wk, const float* __restrict__ bk,
                                const float* __restrict__ wv, const float* __restrict__ bv,
                                float* __restrict__ qo, float* __restrict__ ko,
                                float* __restrict__ vo)
{
    const int wave = (blockIdx.x * blockDim.x + threadIdx.x) >> 5;
    const int lane = threadIdx.x & 31;
    const int tilesPerMat = (ROWS / 16) * (IND / 16);   // 512*32
    const int mat  = wave / tilesPerMat;
    const int rem  = wave % tilesPerMat;
    const int rowt = rem >> 5;
    const int colt = rem & 31;

    const float* W = (mat == 0) ? wq : (mat == 1) ? wk : wv;
    const float* B = (mat == 0) ? bq : (mat == 1) ? bk : bv;
    float*       Y = (mat == 0) ? qo : (mat == 1) ? ko : vo;

    const int half = lane >> 4;
    const int lid  = lane & 15;

    const float* xrow = x + (size_t)(rowt * 16 + lid) * IND + half * 2;
    const float* wrow = W + (size_t)(colt * 16 + lid) * IND + half * 2;

    v8f acc = {};
#pragma unroll 8
    for (int kk = 0; kk < IND; kk += 4) {
        v2f a = *(const v2f*)(xrow + kk);
        v2f b = *(const v2f*)(wrow + kk);
        acc = __builtin_amdgcn_wmma_f32_16x16x4_f32(false, a, false, b,
                                                    (short)0, acc, false, false);
    }

    const float bias = B[colt * 16 + lid];
    float* yb = Y + colt * 16 + lid;
#pragma unroll
    for (int i = 0; i < 8; ++i) {
        const int r = rowt * 16 + i + 8 * half;
        yb[(size_t)r * IND] = acc[i] + bias;
    }
}

// ---------------------------------------------------------------------------
// Kernel 2 (fused attention): block = (n, 16-row q-tile); wave wv = head h.
//  Per K-tile the block async-stages the full 16x512 K (or V) row tile into
//  LDS once; all 8 head-waves consume their interleaved column slices.
//  Pass 1: S = QK^T/8 -> online row max m / sum l (no global writes).
//  Pass 2: recompute S from LDS, emit p = exp(s-m)/l to the attn output
//          (8 waves fill interleaved h-slots of the same cachelines),
//          transpose P through wave-private LDS, accumulate ctx += P @ V.
// ---------------------------------------------------------------------------
__global__ void mha_attn_fused_kernel(const float* __restrict__ Q,
                                      const float* __restrict__ K,
                                      const float* __restrict__ V,
                                      float* __restrict__ attn,
                                      float* __restrict__ ctx)
{
    __shared__ float tile[16][TPAD];     // staged K or V row tile (33KB)
    __shared__ float plds[8][16][17];    // per-wave 16x16 P tile (8.7KB)

    const int wv   = threadIdx.x >> 5;   // wave in block == head h
    const int lane = threadIdx.x & 31;
    const int qt = blockIdx.x & 63;      // 64 q-tiles of 16
    const int n  = blockIdx.x >> 6;
    const int h  = wv;
    const int half = lane >> 4, lid = lane & 15;
    const float scale = 0.125f;          // 1/sqrt(64)

    const unsigned ldsTile0 = (unsigned)(uintptr_t)&tile[0][0];
    const unsigned ldsThreadBase =
        ldsTile0 + (unsigned)(((threadIdx.x >> 7) * TPAD + (threadIdx.x & 127) * 4) * 4);

    const float* Kb = K + (size_t)n * SEQ * IND;
    const float* Vb = V + (size_t)n * SEQ * IND;

    // ---- preload Q tile (16 q x 64 d for head h) into registers ---------
    const float* qbase = Q + (size_t)(n * SEQ + qt * 16 + lid) * IND + h;
    float qa[32];
#pragma unroll
    for (int t = 0; t < 16; ++t) {
        const int d0 = t * 4 + half * 2;
        qa[2 * t]     = qbase[(size_t)d0 * NH];
        qa[2 * t + 1] = qbase[(size_t)(d0 + 1) * NH];
    }

    // ---- pass 1: softmax statistics -------------------------------------
    float m_run[8], l_run[8];
#pragma unroll
    for (int i = 0; i < 8; ++i) { m_run[i] = -1e30f; l_run[i] = 0.0f; }

    for (int kt = 0; kt < 64; ++kt) {
        stage_tile_async(Kb + (size_t)kt * 16 * IND, ldsThreadBase);
        wait_async0();
        __syncthreads();

        v8f acc = {};
#pragma unroll
        for (int t = 0; t < 16; ++t) {
            const int d0 = t * 4 + half * 2;
            v2f a; a.x = qa[2 * t]; a.y = qa[2 * t + 1];
            v2f b; b.x = tile[lid][d0 * NH + h];
            b.y = tile[lid][(d0 + 1) * NH + h];
            acc = __builtin_amdgcn_wmma_f32_16x16x4_f32(false, a, false, b,
                                                        (short)0, acc, false, false);
        }
        __syncthreads();                 // done reading K tile

#pragma unroll
        for (int i = 0; i < 8; ++i) {
            const float s = acc[i] * scale;
            float mx = s;
            mx = fmaxf(mx, __shfl_xor(mx, 1, 32));
            mx = fmaxf(mx, __shfl_xor(mx, 2, 32));
            mx = fmaxf(mx, __shfl_xor(mx, 4, 32));
            mx = fmaxf(mx, __shfl_xor(mx, 8, 32));
            const float mnew = fmaxf(m_run[i], mx);
            float e = __expf(s - mnew);
            e += __shfl_xor(e, 1, 32);
            e += __shfl_xor(e, 2, 32);
            e += __shfl_xor(e, 4, 32);
            e += __shfl_xor(e, 8, 32);
            l_run[i] = l_run[i] * __expf(m_run[i] - mnew) + e;
            m_run[i] = mnew;
        }
    }

    // ---- pass 2: emit attn + accumulate context -------------------------
    v8f cacc[4];
#pragma unroll
    for (int dt = 0; dt < 4; ++dt) cacc[dt] = (v8f){};

    for (int kt = 0; kt < 64; ++kt) {
        // stage K tile, recompute S
        stage_tile_async(Kb + (size_t)kt * 16 * IND, ldsThreadBase);
        wait_async0();
        __syncthreads();

        v8f acc = {};
#pragma unroll
        for (int t = 0; t < 16; ++t) {
            const int d0 = t * 4 + half * 2;
            v2f a; a.x = qa[2 * t]; a.y = qa[2 * t + 1];
            v2f b; b.x = tile[lid][d0 * NH + h];
            b.y = tile[lid][(d0 + 1) * NH + h];
            acc = __builtin_amdgcn_wmma_f32_16x16x4_f32(false, a, false, b,
                                                        (short)0, acc, false, false);
        }
        __syncthreads();                 // done reading K tile

        // stage V tile into the same buffer
        stage_tile_async(Vb + (size_t)kt * 16 * IND, ldsThreadBase);

        // normalized probabilities: write out + stage P in wave-private LDS
#pragma unroll
        for (int i = 0; i < 8; ++i) {
            const int qrow = i + 8 * half;
            const float p = __expf(acc[i] * scale - m_run[i]) / l_run[i];
            attn[(((size_t)(n * SEQ + qt * 16 + qrow)) * SEQ + kt * 16 + lid) * NH + h] = p;
            plds[wv][qrow][lid] = p;
        }

        wait_async0();
        __syncthreads();                 // V tile visible to all waves

        // P @ V: A = P[q][k-chunk] (transposed via LDS), B = V slice
#pragma unroll
        for (int c = 0; c < 4; ++c) {
            const int k0 = c * 4 + half * 2;
            v2f a; a.x = plds[wv][lid][k0]; a.y = plds[wv][lid][k0 + 1];
#pragma unroll
            for (int dt = 0; dt < 4; ++dt) {
                v2f b;
                b.x = tile[k0][(dt * 16 + lid) * NH + h];
                b.y = tile[k0 + 1][(dt * 16 + lid) * NH + h];
                cacc[dt] = __builtin_amdgcn_wmma_f32_16x16x4_f32(false, a, false, b,
                                                                 (short)0, cacc[dt],
                                                                 false, false);
            }
        }
        __syncthreads();                 // done reading V before next stage
    }

    // ---- store context ---------------------------------------------------
#pragma unroll
    for (int dt = 0; dt < 4; ++dt) {
#pragma unroll
        for (int i = 0; i < 8; ++i) {
            const int q = qt * 16 + i + 8 * half;
            const int d = dt * 16 + lid;
            ctx[((size_t)(n * SEQ + q)) * IND + d * NH + h] = cacc[dt][i];
        }
    }
}

// ---------------------------------------------------------------------------
extern "C" void kernel_launch(void* const* d_in, const int* in_sizes, int n_in,
                              void* d_out, int out_size, void* d_ws, size_t ws_size,
                              hipStream_t stream)
{
    const float* x  = (const float*)d_in[0];
    const float* wq = (const float*)d_in[1];
    const float* bq = (const float*)d_in[2];
    const float* wk = (const float*)d_in[3];
    const float* bk = (const float*)d_in[4];
    const float* wv = (const float*)d_in[5];
    const float* bv = (const float*)d_in[6];

    float* ctx  = (float*)d_out;                         // [8,1024,512]
    float* attn = ctx + (size_t)NB * SEQ * IND;          // [8,1024,1024,8]

    float* Qws  = (float*)d_ws;                          // [8192,512]
    float* Kws  = Qws + (size_t)ROWS * IND;
    float* Vws  = Kws + (size_t)ROWS * IND;

    // 1) projections: 3 * 512 * 32 tiles, 8 waves (256 thr) per block
    const int projTiles = 3 * (ROWS / 16) * (IND / 16);
    mha_proj_kernel<<<projTiles / 8, 256, 0, stream>>>(x, wq, bq, wk, bk, wv, bv,
                                                       Qws, Kws, Vws);

    // 2) fused attention: block = (n, q-tile), wave = head; 512 blocks
    mha_attn_fused_kernel<<<NB * (SEQ / 16), 256, 0, stream>>>(Qws, Kws, Vws,
                                                               attn, ctx);
}